// SumRGGCN_40699110097748
// MI455X (gfx1250) — compile-verified
//
#include <hip/hip_runtime.h>

#define N_USERS 60000
#define N_ITEMS 40000
#define N_NODES 100000
#define N_EDGES 1000000
#define D 64
#define BN_EPS 1e-3f

typedef float v2f __attribute__((ext_vector_type(2)));
typedef float v8f __attribute__((ext_vector_type(8)));

// ---------------------------------------------------------------------------
// concat: x0 = [u_features ; v_features]
// ---------------------------------------------------------------------------
__global__ __launch_bounds__(256) void concat_kernel(const float* __restrict__ u,
                                                     const float* __restrict__ v,
                                                     float* __restrict__ x0) {
    int idx = blockIdx.x * 256 + threadIdx.x;
    const int nu = N_USERS * D;
    const int tot = N_NODES * D;
    if (idx >= tot) return;
    x0[idx] = (idx < nu) ? u[idx] : v[idx - nu];
}

// ---------------------------------------------------------------------------
// Y[nrows,64] = X[nrows,64] @ W[64,64] (+ bias), via V_WMMA_F32_16X16X4_F32.
// One wave computes a 16x16 output tile; K=64 -> 16 chained WMMAs per tile.
// A layout (16x4 f32): lanes 0-15 hold M=0..15 with K = {k0, k0+1} in the two
// VGPRs; lanes 16-31 hold K = {k0+2, k0+3}.  B mirrors with N across lanes.
// C/D: VGPR r -> M=r (lanes 0-15) / M=r+8 (lanes 16-31), N = lane%16.
// ---------------------------------------------------------------------------
__global__ __launch_bounds__(256) void gemm64_wmma(const float* __restrict__ X,
                                                   const float* __restrict__ W,
                                                   const float* __restrict__ bias,
                                                   float* __restrict__ Y,
                                                   int nrows) {
    __shared__ float sW[64 * 64];
    for (int i = threadIdx.x; i < 64 * 64; i += 256) sW[i] = W[i];
    __syncthreads();

    const int lane = threadIdx.x & 31;
    const int wv   = threadIdx.x >> 5;
    const int m0   = (blockIdx.x * 8 + wv) * 16;
    if (m0 >= nrows) return;               // whole-wave guard: EXEC stays all-1s

    const int l15   = lane & 15;
    const int khalf = (lane >> 4) << 1;    // 0 for lanes 0-15, 2 for lanes 16-31
    const float* xrow = X + (size_t)(m0 + l15) * D;

    // Preload this lane's A fragments for the whole K=64 sweep (16 x float2).
    v2f areg[16];
#pragma unroll
    for (int kk = 0; kk < 16; ++kk)
        areg[kk] = *(const v2f*)(xrow + kk * 4 + khalf);

#pragma unroll
    for (int n0 = 0; n0 < 64; n0 += 16) {
        v8f c = {};
#pragma unroll
        for (int kk = 0; kk < 16; ++kk) {
            const int k0 = kk * 4;
            v2f b;
            b.x = sW[(k0 + khalf) * 64 + n0 + l15];
            b.y = sW[(k0 + khalf + 1) * 64 + n0 + l15];
            c = __builtin_amdgcn_wmma_f32_16x16x4_f32(
                    false, areg[kk], false, b, (short)0, c, false, false);
        }
        const int col   = n0 + l15;
        const float bb  = bias ? bias[col] : 0.0f;
        const int rbase = m0 + ((lane >> 4) << 3);
#pragma unroll
        for (int r = 0; r < 8; ++r)
            Y[(size_t)(rbase + r) * D + col] = c[r] + bb;
    }
}

// ---------------------------------------------------------------------------
// Edge phase: one wave per edge, 2 columns per lane (float2, coalesced rows).
//   gate = sigmoid(pVi[e_end] + pVj[e_start] + bv)
//   acc[e_end] += gate * pUi[e_start]          (global_atomic_add_f32)
// acc was pre-initialized with X@Uj + bu by gemm64_wmma.
// ---------------------------------------------------------------------------
__global__ __launch_bounds__(256) void edge_kernel(const int* __restrict__ es,
                                                   const int* __restrict__ ee,
                                                   const float* __restrict__ pUi,
                                                   const float* __restrict__ pVi,
                                                   const float* __restrict__ pVj,
                                                   const float* __restrict__ bv,
                                                   float* __restrict__ acc) {
    const int lane = threadIdx.x & 31;
    const int wv   = threadIdx.x >> 5;
    const int e    = blockIdx.x * 8 + wv;
    if (e >= N_EDGES) return;
    const int s = es[e];
    const int d = ee[e];

    const float2 vi = ((const float2*)(pVi + (size_t)d * D))[lane];
    const float2 vj = ((const float2*)(pVj + (size_t)s * D))[lane];
    const float2 ui = ((const float2*)(pUi + (size_t)s * D))[lane];
    const float2 bb = ((const float2*)bv)[lane];

    const float gx = 1.0f / (1.0f + __expf(-(vi.x + vj.x + bb.x)));
    const float gy = 1.0f / (1.0f + __expf(-(vi.y + vj.y + bb.y)));

    float* dst = acc + (size_t)d * D + lane * 2;
    atomicAdd(dst + 0, gx * ui.x);
    atomicAdd(dst + 1, gy * ui.y);
}

// ---------------------------------------------------------------------------
// Per-column sum / sumsq reduction (for batchnorm statistics).
// stats[0..63] = sum, stats[64..127] = sumsq.  Must be zeroed beforehand.
// ---------------------------------------------------------------------------
__global__ __launch_bounds__(256) void zero128(float* __restrict__ p) {
    p[threadIdx.x] = 0.0f;
}

__global__ __launch_bounds__(256) void colstats_kernel(const float* __restrict__ X,
                                                       float* __restrict__ stats) {
    __shared__ float ssum[64];
    __shared__ float ssq[64];
    if (threadIdx.x < 64) { ssum[threadIdx.x] = 0.0f; ssq[threadIdx.x] = 0.0f; }
    __syncthreads();

    const int col = threadIdx.x & 63;
    const int r0  = blockIdx.x * 4 + (threadIdx.x >> 6);
    const int rs  = gridDim.x * 4;
    float s = 0.0f, q = 0.0f;
    for (int r = r0; r < N_NODES; r += rs) {
        const float v = X[(size_t)r * D + col];
        s += v;
        q += v * v;
    }
    atomicAdd(&ssum[col], s);
    atomicAdd(&ssq[col], q);
    __syncthreads();
    if (threadIdx.x < 64) {
        atomicAdd(&stats[threadIdx.x], ssum[threadIdx.x]);
        atomicAdd(&stats[64 + threadIdx.x], ssq[threadIdx.x]);
    }
}

// ---------------------------------------------------------------------------
// Fused batchnorm apply passes.
// ---------------------------------------------------------------------------
__global__ __launch_bounds__(256) void bn_relu_kernel(const float* __restrict__ X,
                                                      const float* __restrict__ stats,
                                                      const float* __restrict__ gamma,
                                                      const float* __restrict__ beta,
                                                      float* __restrict__ Y) {
    const int idx = blockIdx.x * 256 + threadIdx.x;
    if (idx >= N_NODES * D) return;
    const int col = threadIdx.x & 63;          // blockDim multiple of 64
    const float inv_n = 1.0f / (float)N_NODES;
    const float mean = stats[col] * inv_n;
    const float var  = stats[64 + col] * inv_n - mean * mean;
    const float inv  = rsqrtf(var + BN_EPS);
    const float v = gamma[col] * (X[idx] - mean) * inv + beta[col];
    Y[idx] = fmaxf(v, 0.0f);
}

__global__ __launch_bounds__(256) void bn_res_relu_kernel(const float* __restrict__ X,
                                                          const float* __restrict__ stats,
                                                          const float* __restrict__ gamma,
                                                          const float* __restrict__ beta,
                                                          const float* __restrict__ tR,
                                                          float* __restrict__ Y) {
    const int idx = blockIdx.x * 256 + threadIdx.x;
    if (idx >= N_NODES * D) return;
    const int col = threadIdx.x & 63;
    const float inv_n = 1.0f / (float)N_NODES;
    const float mean = stats[col] * inv_n;
    const float var  = stats[64 + col] * inv_n - mean * mean;
    const float inv  = rsqrtf(var + BN_EPS);
    const float v = gamma[col] * (X[idx] - mean) * inv + beta[col] + tR[idx];
    Y[idx] = fmaxf(v, 0.0f);
}

// ---------------------------------------------------------------------------
// Host-side orchestration
// ---------------------------------------------------------------------------
extern "C" void kernel_launch(void* const* d_in, const int* in_sizes, int n_in,
                              void* d_out, int out_size, void* d_ws, size_t ws_size,
                              hipStream_t stream) {
    const float* u_feat = (const float*)d_in[0];
    const float* v_feat = (const float*)d_in[1];
    const int*   es     = (const int*)d_in[2];
    const int*   ee     = (const int*)d_in[3];
    const float* Ui1 = (const float*)d_in[4];
    const float* Uj1 = (const float*)d_in[5];
    const float* Vi1 = (const float*)d_in[6];
    const float* Vj1 = (const float*)d_in[7];
    const float* bu1 = (const float*)d_in[8];
    const float* bv1 = (const float*)d_in[9];
    const float* Ui2 = (const float*)d_in[10];
    const float* Uj2 = (const float*)d_in[11];
    const float* Vi2 = (const float*)d_in[12];
    const float* Vj2 = (const float*)d_in[13];
    const float* bu2 = (const float*)d_in[14];
    const float* bv2 = (const float*)d_in[15];
    const float* Rw  = (const float*)d_in[16];
    const float* g1  = (const float*)d_in[17];
    const float* b1  = (const float*)d_in[18];
    const float* g2  = (const float*)d_in[19];
    const float* b2  = (const float*)d_in[20];
    float* out = (float*)d_out;

    float* ws = (float*)d_ws;
    const size_t S = (size_t)N_NODES * D;
    float* x0    = ws + 0 * S;
    float* Bx    = ws + 1 * S;   // stage-1 post-BN activations
    float* pUi   = ws + 2 * S;   // also reused for x0@R residual branch
    float* pVi   = ws + 3 * S;
    float* pVj   = ws + 4 * S;
    float* acc   = ws + 5 * S;   // X@Uj + bu, then scatter-accumulated
    float* stats = ws + 6 * S;   // 128 floats

    const int elem_blocks = (N_NODES * D + 255) / 256;          // 25000
    const int gemm_blocks = (N_NODES + 127) / 128;              // 782
    const int edge_blocks = (N_EDGES + 7) / 8;                  // 125000
    const int stat_blocks = 512;

    concat_kernel<<<elem_blocks, 256, 0, stream>>>(u_feat, v_feat, x0);

    // ---- stage 1 ----
    gemm64_wmma<<<gemm_blocks, 256, 0, stream>>>(x0, Ui1, nullptr, pUi, N_NODES);
    gemm64_wmma<<<gemm_blocks, 256, 0, stream>>>(x0, Vi1, nullptr, pVi, N_NODES);
    gemm64_wmma<<<gemm_blocks, 256, 0, stream>>>(x0, Vj1, nullptr, pVj, N_NODES);
    gemm64_wmma<<<gemm_blocks, 256, 0, stream>>>(x0, Uj1, bu1,     acc, N_NODES);
    edge_kernel<<<edge_blocks, 256, 0, stream>>>(es, ee, pUi, pVi, pVj, bv1, acc);
    zero128<<<1, 128, 0, stream>>>(stats);
    colstats_kernel<<<stat_blocks, 256, 0, stream>>>(acc, stats);
    bn_relu_kernel<<<elem_blocks, 256, 0, stream>>>(acc, stats, g1, b1, Bx);

    // ---- stage 2 ----
    gemm64_wmma<<<gemm_blocks, 256, 0, stream>>>(Bx, Ui2, nullptr, pUi, N_NODES);
    gemm64_wmma<<<gemm_blocks, 256, 0, stream>>>(Bx, Vi2, nullptr, pVi, N_NODES);
    gemm64_wmma<<<gemm_blocks, 256, 0, stream>>>(Bx, Vj2, nullptr, pVj, N_NODES);
    gemm64_wmma<<<gemm_blocks, 256, 0, stream>>>(Bx, Uj2, bu2,     acc, N_NODES);
    edge_kernel<<<edge_blocks, 256, 0, stream>>>(es, ee, pUi, pVi, pVj, bv2, acc);

    // residual branch: pUi (now free) <- x0 @ R
    gemm64_wmma<<<gemm_blocks, 256, 0, stream>>>(x0, Rw, nullptr, pUi, N_NODES);

    zero128<<<1, 128, 0, stream>>>(stats);
    colstats_kernel<<<stat_blocks, 256, 0, stream>>>(acc, stats);
    bn_res_relu_kernel<<<elem_blocks, 256, 0, stream>>>(acc, stats, g2, b2, pUi, out);
}